// SAGEConv_29781303231102
// MI455X (gfx1250) — compile-verified
//
#include <hip/hip_runtime.h>

// CDNA5 / gfx1250, wave32. Fused SAGEConv + ReLU + BatchNorm + residual.
// Matrix ops use V_WMMA_F32_16X16X4_F32 (exact fp32 WMMA).

typedef __attribute__((ext_vector_type(2))) float v2f;
typedef __attribute__((ext_vector_type(8))) float v8f;

#define FEAT 64
#define BN_EPS 1e-5f

// ---------------------------------------------------------------- zero scratch
__global__ void zero_ws_kernel(float* __restrict__ ws, long long n) {
  long long i = (long long)blockIdx.x * blockDim.x + threadIdx.x;
  if (i < n) ws[i] = 0.0f;
}

// ------------------------------------------------- edge scatter (mean aggregation)
// 16 lanes per edge; each lane handles 4 consecutive features (float4 gather,
// 4 hardware fp32 atomics into L2 — everything is L2-resident on MI455X).
__global__ void scatter_kernel(const float* __restrict__ x, const int* __restrict__ ei,
                               float* __restrict__ agg, float* __restrict__ cnt, int E) {
  long long tid = (long long)blockIdx.x * blockDim.x + threadIdx.x;
  int e    = (int)(tid >> 4);
  int lane = (int)(tid & 15);
  if (e >= E) return;
  int src = ei[e];
  int dst = ei[E + e];
  const float4 v = *(const float4*)(x + (size_t)src * FEAT + lane * 4);
  float* a = agg + (size_t)dst * FEAT + lane * 4;
  __hip_atomic_fetch_add(a + 0, v.x, __ATOMIC_RELAXED, __HIP_MEMORY_SCOPE_AGENT);
  __hip_atomic_fetch_add(a + 1, v.y, __ATOMIC_RELAXED, __HIP_MEMORY_SCOPE_AGENT);
  __hip_atomic_fetch_add(a + 2, v.z, __ATOMIC_RELAXED, __HIP_MEMORY_SCOPE_AGENT);
  __hip_atomic_fetch_add(a + 3, v.w, __ATOMIC_RELAXED, __HIP_MEMORY_SCOPE_AGENT);
  if (lane == 0)
    __hip_atomic_fetch_add(cnt + dst, 1.0f, __ATOMIC_RELAXED, __HIP_MEMORY_SCOPE_AGENT);
}

// ------------------------------------- fused WMMA GEMM + bias + ReLU + BN stats
// block = 128 threads = 4 waves; wave w owns output columns [16w, 16w+16).
// Grid-stride over 16-row tiles; B fragments (Wl^T, Wr^T) live in registers.
__global__ void sage_wmma_kernel(const float* __restrict__ x, const float* __restrict__ agg,
                                 const float* __restrict__ cnt, const float* __restrict__ Wl,
                                 const float* __restrict__ bl, const float* __restrict__ Wr,
                                 float* __restrict__ h, float* __restrict__ sums,
                                 float* __restrict__ sumsq, int nTiles, int N) {
  const int wave = threadIdx.x >> 5;
  const int lane = threadIdx.x & 31;
  const int half = lane >> 4;      // 0: lanes 0-15, 1: lanes 16-31
  const int ln   = lane & 15;
  const int col  = wave * 16 + ln; // output column this lane owns (B/C/D N-dim)

  // B fragments: B[k, j] = W[j*64 + k]  (W row-major [64,64], we need W^T)
  // Lane layout for 16x16x4 f32 B: lanes<16 hold K=4kk,4kk+1; lanes>=16 hold K=4kk+2,4kk+3.
  v2f Bl[16], Br[16];
#pragma unroll
  for (int kk = 0; kk < 16; ++kk) {
    int k = kk * 4 + half * 2;
    Bl[kk] = *(const v2f*)(Wl + (size_t)col * FEAT + k);
    Br[kk] = *(const v2f*)(Wr + (size_t)col * FEAT + k);
  }
  const float bias = bl[col];

  float csum = 0.0f, csq = 0.0f;  // per-column partial BN stats

  for (int tile = blockIdx.x; tile < nTiles; tile += gridDim.x) {
    const int row0 = tile * 16;
    int rowA = row0 + ln;                      // A-matrix row this lane loads
    if (rowA >= N) rowA = N - 1;               // clamp (EXEC must stay all-1s)
    const float inv = 1.0f / fmaxf(cnt[rowA], 1.0f);

    v8f acc;
#pragma unroll
    for (int v = 0; v < 8; ++v) acc[v] = bias; // start from broadcast bias

    // h += mean_nb @ Wl^T : A = agg * inv
    const float* aggRow = agg + (size_t)rowA * FEAT + half * 2;
#pragma unroll
    for (int kk = 0; kk < 16; ++kk) {
      v2f a = *(const v2f*)(aggRow + kk * 4);
      a[0] *= inv; a[1] *= inv;
      acc = __builtin_amdgcn_wmma_f32_16x16x4_f32(false, a, false, Bl[kk],
                                                  (short)0, acc, false, false);
    }
    // h += x @ Wr^T
    const float* xRow = x + (size_t)rowA * FEAT + half * 2;
#pragma unroll
    for (int kk = 0; kk < 16; ++kk) {
      v2f a = *(const v2f*)(xRow + kk * 4);
      acc = __builtin_amdgcn_wmma_f32_16x16x4_f32(false, a, false, Br[kk],
                                                  (short)0, acc, false, false);
    }

    // ReLU + store h tile + accumulate BN partial sums.
    // C/D layout: VGPR v -> row row0+v (lanes 0-15) or row0+v+8 (lanes 16-31).
#pragma unroll
    for (int v = 0; v < 8; ++v) {
      int row = row0 + v + half * 8;
      if (row < N) {
        float hv = fmaxf(acc[v], 0.0f);
        h[(size_t)row * FEAT + col] = hv;
        csum += hv;
        csq  += hv * hv;
      }
    }
  }

  // lanes n and n+16 hold the same column: fold halves, then one atomic/column.
  csum += __shfl_xor(csum, 16, 32);
  csq  += __shfl_xor(csq, 16, 32);
  if (half == 0) {
    __hip_atomic_fetch_add(&sums[col],  csum, __ATOMIC_RELAXED, __HIP_MEMORY_SCOPE_AGENT);
    __hip_atomic_fetch_add(&sumsq[col], csq,  __ATOMIC_RELAXED, __HIP_MEMORY_SCOPE_AGENT);
  }
}

// ------------------------------------------------ BN normalize + residual add
__global__ void finalize_kernel(const float* __restrict__ x, const float* __restrict__ h,
                                const float* __restrict__ sums, const float* __restrict__ sumsq,
                                const float* __restrict__ gamma, const float* __restrict__ beta,
                                float* __restrict__ out, int N) {
  long long i = (long long)blockIdx.x * blockDim.x + threadIdx.x;
  long long total = (long long)N * FEAT;
  if (i >= total) return;
  int d = (int)(i & (FEAT - 1));
  float invN = 1.0f / (float)N;
  float mu   = sums[d] * invN;
  float var  = sumsq[d] * invN - mu * mu;         // biased variance = E[h^2]-mu^2
  float rs   = rsqrtf(var + BN_EPS);
  out[i] = x[i] + (h[i] - mu) * rs * gamma[d] + beta[d];
}

// ----------------------------------------------------------------------------
extern "C" void kernel_launch(void* const* d_in, const int* in_sizes, int n_in,
                              void* d_out, int out_size, void* d_ws, size_t ws_size,
                              hipStream_t stream) {
  const float* x     = (const float*)d_in[0];
  const int*   ei    = (const int*)d_in[1];
  const float* Wl    = (const float*)d_in[2];
  const float* bl    = (const float*)d_in[3];
  const float* Wr    = (const float*)d_in[4];
  const float* gamma = (const float*)d_in[5];
  const float* beta  = (const float*)d_in[6];

  const int N = in_sizes[0] / FEAT;
  const int E = in_sizes[1] / 2;

  // Workspace layout (floats): agg[N*64] | cnt[N] | sums[64] | sumsq[64] | h[N*64]
  float* agg   = (float*)d_ws;
  float* cnt   = agg + (size_t)N * FEAT;
  float* sums  = cnt + N;
  float* sumsq = sums + FEAT;
  float* h     = sumsq + FEAT;

  long long nzero = (long long)N * FEAT + N + 2 * FEAT;
  zero_ws_kernel<<<(int)((nzero + 255) / 256), 256, 0, stream>>>((float*)d_ws, nzero);

  long long nscatter = (long long)E * 16;
  scatter_kernel<<<(int)((nscatter + 255) / 256), 256, 0, stream>>>(x, ei, agg, cnt, E);

  int nTiles = (N + 15) / 16;
  int gemmBlocks = nTiles < 1024 ? nTiles : 1024;
  sage_wmma_kernel<<<gemmBlocks, 128, 0, stream>>>(x, agg, cnt, Wl, bl, Wr,
                                                   h, sums, sumsq, nTiles, N);

  long long ntot = (long long)N * FEAT;
  finalize_kernel<<<(int)((ntot + 255) / 256), 256, 0, stream>>>(x, h, sums, sumsq,
                                                                 gamma, beta,
                                                                 (float*)d_out, N);
}